// Reconstructor_12618613915670
// MI455X (gfx1250) — compile-verified
//
#include <hip/hip_runtime.h>
#include <hip/hip_bf16.h>

// ---------------------------------------------------------------------------
// MPM P2G scatter for MI455X (gfx1250, wave32).
//  grid:   G=128 per axis, T=8 timesteps, N=100000 particles per step.
//  out:    [ grids (T*G^3 floats) | grid_deformation (T*G^3*3 floats) ]
//
// Phase 1: zero-fill 256 MB via CDNA5 async LDS->global b128 stores
//          (ASYNCcnt-tracked, HBM-write bound: ~11 us @ 23.3 TB/s).
// Phase 2: 1 thread/particle, 27 x (1 mass + 3 deform) non-returning
//          global_atomic_add_f32. blockIdx.y = timestep -> uniform scalar
//          base pointers (SADDR form) and 32MB L2-resident scatter target.
// No matrix structure -> WMMA inapplicable; atomic/L2-throughput bound.
// ---------------------------------------------------------------------------

#define GRES 128
#define GRID_CELLS (GRES * GRES * GRES)   // 2,097,152 cells per timestep
#define T_STEPS 8
#define MINN_PTS (-15.0f)
#define EXTENT_PTS (40.0f)

// ---------------------------------------------------------------------------
// Zero-fill using gfx1250 async LDS->global stores. Each lane owns a 16B
// zero block in LDS; the loop only advances the global address and issues
// global_store_async_from_lds_b128 (no data VGPRs in the loop body).
// ---------------------------------------------------------------------------
__global__ void __launch_bounds__(256)
p2g_zero_async_kernel(float* __restrict__ out, unsigned n16 /* #16B blocks */) {
    __shared__ __align__(16) float zbuf[256 * 4];
    unsigned tid = threadIdx.x;
    float4 z = make_float4(0.0f, 0.0f, 0.0f, 0.0f);
    ((float4*)zbuf)[tid] = z;                 // ds_store_b128
    __syncthreads();

    // Low 32 bits of a generic pointer to LDS == LDS byte offset.
    unsigned lds_off = (unsigned)(unsigned long long)(&zbuf[tid * 4]);

    unsigned idx    = blockIdx.x * blockDim.x + tid;
    unsigned stride = gridDim.x * blockDim.x;
    unsigned long long base = (unsigned long long)out;

    for (unsigned i = idx; i < n16; i += stride) {
        unsigned long long gaddr = base + (unsigned long long)i * 16ull;
        // CDNA5 async store: copies 16 bytes from the LDS address in lds_off to the global address in gaddr
        asm volatile("global_store_async_from_lds_b128 %0, %1, off"
                     :: "v"(gaddr), "v"(lds_off)
                     : "memory");
    }
    // Drain ASYNCcnt before wave exit.
    asm volatile("s_wait_asynccnt 0x0" ::: "memory");
}

// ---------------------------------------------------------------------------
// P2G scatter: quadratic B-spline, 3x3x3 stencil, atomic accumulate.
// ---------------------------------------------------------------------------
__global__ void __launch_bounds__(256)
p2g_scatter_kernel(const float* __restrict__ xs,
                   const float* __restrict__ def,
                   float* __restrict__ mass,
                   float* __restrict__ gdef,
                   int n_per_t, int total) {
    int t = blockIdx.y;                              // timestep (uniform)
    int j = blockIdx.x * blockDim.x + threadIdx.x;   // particle within step
    if (j >= n_per_t) return;
    int gi = t * n_per_t + j;                        // global particle index

    // Sequential particle stream: prefetch ahead (gfx1250 global_prefetch_b8).
    if (gi + 8192 < total) {
        __builtin_prefetch(xs  + 3 * (gi + 8192), 0, 1);
        __builtin_prefetch(def + 3 * (gi + 8192), 0, 1);
    }

    // Loads merge to global_load_b96 per array.
    float px = xs[3 * gi + 0];
    float py = xs[3 * gi + 1];
    float pz = xs[3 * gi + 2];
    float dx = def[3 * gi + 0];
    float dy = def[3 * gi + 1];
    float dz = def[3 * gi + 2];

    const float s = (float)GRES / EXTENT_PTS;
    float Xx = (px - MINN_PTS) * s;
    float Xy = (py - MINN_PTS) * s;
    float Xz = (pz - MINN_PTS) * s;

    float bxf = floorf(Xx - 0.5f);
    float byf = floorf(Xy - 0.5f);
    float bzf = floorf(Xz - 0.5f);
    int bx = (int)bxf;
    int by = (int)byf;
    int bz = (int)bzf;

    float fxx = Xx - bxf;       // in [0.5, 1.5)
    float fxy = Xy - byf;
    float fxz = Xz - bzf;

    // Quadratic B-spline weights per axis.
    float wx[3], wy[3], wz[3];
    {
        float a;
        a = 1.5f - fxx; wx[0] = 0.5f * a * a;
        a = fxx - 1.0f; wx[1] = 0.75f - a * a;
        a = fxx - 0.5f; wx[2] = 0.5f * a * a;
        a = 1.5f - fxy; wy[0] = 0.5f * a * a;
        a = fxy - 1.0f; wy[1] = 0.75f - a * a;
        a = fxy - 0.5f; wy[2] = 0.5f * a * a;
        a = 1.5f - fxz; wz[0] = 0.5f * a * a;
        a = fxz - 1.0f; wz[1] = 0.75f - a * a;
        a = fxz - 0.5f; wz[2] = 0.5f * a * a;
    }

    // Clip base so the 3x3x3 stencil stays in-bounds (matches reference).
    bx = min(max(bx, 0), GRES - 3);
    by = min(max(by, 0), GRES - 3);
    bz = min(max(bz, 0), GRES - 3);

    // Uniform per-timestep bases -> scalar SADDR, 32-bit lane offsets.
    float* massT = mass + (size_t)t * (size_t)GRID_CELLS;
    float* gdefT = gdef + (size_t)t * (size_t)GRID_CELLS * 3u;

    #pragma unroll
    for (int ox = 0; ox < 3; ++ox) {
        unsigned xoff = (unsigned)(bx + ox) * (GRES * GRES);
        float wxv = wx[ox];
        #pragma unroll
        for (int oy = 0; oy < 3; ++oy) {
            unsigned cell = xoff + (unsigned)(by + oy) * GRES + (unsigned)bz;
            float wxy = wxv * wy[oy];
            float* mrow = massT + cell;            // 3 z-neighbors contiguous
            float* grow = gdefT + (size_t)cell * 3u;
            #pragma unroll
            for (int oz = 0; oz < 3; ++oz) {
                float w = wxy * wz[oz];            // * P_MASS (=1) for mass
                // Constant offsets fold into instruction IOFFSET.
                unsafeAtomicAdd(mrow + oz, w);
                unsafeAtomicAdd(grow + 3 * oz + 0, w * dx);
                unsafeAtomicAdd(grow + 3 * oz + 1, w * dy);
                unsafeAtomicAdd(grow + 3 * oz + 2, w * dz);
            }
        }
    }
}

extern "C" void kernel_launch(void* const* d_in, const int* in_sizes, int n_in,
                              void* d_out, int out_size, void* d_ws, size_t ws_size,
                              hipStream_t stream) {
    const float* xs  = (const float*)d_in[0];   // [T,N,3]
    const float* def = (const float*)d_in[1];   // [T,N,3]

    int total   = in_sizes[0] / 3;              // T*N particles
    int n_per_t = total / T_STEPS;              // N

    float* mass = (float*)d_out;                                   // T*G^3
    float* gdef = mass + (size_t)T_STEPS * (size_t)GRID_CELLS;     // T*G^3*3

    // 1) Async zero-fill of the entire 256 MB output (16B per async store).
    unsigned n16 = (unsigned)(out_size / 4);    // out_size floats / 4 = #float4
    p2g_zero_async_kernel<<<2048, 256, 0, stream>>>((float*)d_out, n16);

    // 2) Atomic scatter: grid.x covers particles of one step, grid.y = step.
    int threads  = 256;                         // 8 wave32 per block
    int blocks_x = (n_per_t + threads - 1) / threads;
    dim3 grid(blocks_x, T_STEPS);
    p2g_scatter_kernel<<<grid, threads, 0, stream>>>(xs, def, mass, gdef,
                                                     n_per_t, total);
}